// PINN_12816182411701
// MI455X (gfx1250) — compile-verified
//
#include <hip/hip_runtime.h>

// Problem constants (from setup_inputs / reference)
#define PADC  32
#define BATCH 16
#define NLAM  128
#define HH    128
#define WW    128
#define HP    192          // 128 + 2*32
#define WP    192
#define NK    4
#define NTILE 12           // 192 / 16

typedef __attribute__((ext_vector_type(2))) float v2f;
typedef __attribute__((ext_vector_type(8))) float v8f;

// ---------------------------------------------------------------------------
// Kernel 1: prep.
//  - 1D separable PSF factor g[j] = sum_i psf[j][i]  (psf == outer(g,g), sum g = 1)
//  - per-(k,lambda) integer offsets and bilinear fractions from dx/dy
// ---------------------------------------------------------------------------
__global__ void prep_kernel(const float* __restrict__ psf,
                            const float* __restrict__ dx,
                            const float* __restrict__ dy,
                            float* __restrict__ g,
                            float* __restrict__ fx, float* __restrict__ fy,
                            int* __restrict__ nx, int* __restrict__ ny) {
    const int t = threadIdx.x;
    if (t < 8) {
        float s = 0.f;
        if (t < 7) {
            for (int i = 0; i < 7; ++i) s += psf[t * 7 + i];
        }
        g[t] = s;                       // g[7] = 0 sentinel for band clamping
    }
    if (t < NK * NLAM) {
        const float vx = dx[t], vy = dy[t];
        const float nfx = floorf(vx), nfy = floorf(vy);
        nx[t] = (int)nfx;  fx[t] = vx - nfx;
        ny[t] = (int)nfy;  fy[t] = vy - nfy;
    }
}

// ---------------------------------------------------------------------------
// Kernel 2: lambda-accumulation (gather form of the uniform bilinear shift).
// out[y,x] (padded coords) = sum_l  fy*fx*C(y-ny-1, x-nx-1) + fy*(1-fx)*C(y-ny-1, x-nx)
//                                 + (1-fy)*fx*C(y-ny, x-nx-1) + (1-fy)*(1-fx)*C(y-ny, x-nx)
// where C() indexes the *unpadded* cube (coords - PAD) with zeros outside.
// One 256-thread block per 16x16 tile of one (b,k) image; lambda params in LDS.
// ---------------------------------------------------------------------------
__global__ void __launch_bounds__(256)
accum_kernel(const float* __restrict__ cube,
             const float* __restrict__ fxg, const float* __restrict__ fyg,
             const int* __restrict__ nxg, const int* __restrict__ nyg,
             float* __restrict__ acc) {
    __shared__ float sfx[NLAM], sfy[NLAM];
    __shared__ int   snx[NLAM], sny[NLAM];

    const int tile = blockIdx.x;            // 0..143
    const int k    = blockIdx.y;            // 0..3
    const int b    = blockIdx.z;            // 0..15
    const int x0   = (tile % NTILE) * 16;
    const int y0   = (tile / NTILE) * 16;
    const int t    = threadIdx.x;

    if (t < NLAM) {
        sfx[t] = fxg[k * NLAM + t];
        sfy[t] = fyg[k * NLAM + t];
        snx[t] = nxg[k * NLAM + t];
        sny[t] = nyg[k * NLAM + t];
    }
    __syncthreads();

    const int x = x0 + (t & 15);
    const int y = y0 + (t >> 4);
    const float* __restrict__ cb = cube + (size_t)b * (NLAM * HH * WW);

    float a = 0.f;
    for (int l = 0; l < NLAM; ++l) {
        const float* __restrict__ c = cb + (size_t)l * (HH * WW);
        const int r1 = y - sny[l] - PADC;   // cube row for weight (1-fy)
        const int c1 = x - snx[l] - PADC;   // cube col for weight (1-fx)
        const int r0 = r1 - 1, c0 = c1 - 1;
        const float wx = sfx[l], wy = sfy[l];
        const float v00 = ((unsigned)r0 < (unsigned)HH && (unsigned)c0 < (unsigned)WW) ? c[r0 * WW + c0] : 0.f;
        const float v01 = ((unsigned)r0 < (unsigned)HH && (unsigned)c1 < (unsigned)WW) ? c[r0 * WW + c1] : 0.f;
        const float v10 = ((unsigned)r1 < (unsigned)HH && (unsigned)c0 < (unsigned)WW) ? c[r1 * WW + c0] : 0.f;
        const float v11 = ((unsigned)r1 < (unsigned)HH && (unsigned)c1 < (unsigned)WW) ? c[r1 * WW + c1] : 0.f;
        a += wy * (wx * v00 + (1.f - wx) * v01)
           + (1.f - wy) * (wx * v10 + (1.f - wx) * v11);
    }
    acc[(((size_t)b * NK + k) * HP + y) * WP + x] = a;
}

// ---------------------------------------------------------------------------
// Kernel 3: separable 7x7 PSF conv via V_WMMA_F32_16X16X4_F32.
// One wave32 per 16x16 output tile. Per tile:
//   stage 22x24 halo window (zero padded) in LDS;
//   pass 1 (horizontal): T(24x16) = In(24x24) * Bg(24x16), Bg[k][x] = g[k-x]
//   pass 2 (vertical):   Out(16x16) = Ag(16x24) * T(24x16), Ag[y][k] = g[k-y]
// Fragment layouts per ISA 7.12.2 (f32): A[m][k] -> lane m+16*(k>=2), vgpr k&1;
// B[k][n] -> lane n+16*(k>=2), vgpr k&1; D[m][n] -> vgpr m%8, lane n+16*(m/8).
// ---------------------------------------------------------------------------
__global__ void __launch_bounds__(32)
conv_wmma_kernel(const float* __restrict__ acc,
                 const float* __restrict__ g7,
                 float* __restrict__ out) {
    __shared__ float sIn[32][25];   // rows 22..31 and cols 22..23 are zeros
    __shared__ float sT[24][17];    // intermediate (rows 22..23 are zeros)
    __shared__ float sG[8];         // g[0..6], sG[7] = 0

    const int tile = blockIdx.x;    // 0..143
    const int img  = blockIdx.y;    // 0..63 (b*4 + k)
    const int x0   = (tile % NTILE) * 16;
    const int y0   = (tile / NTILE) * 16;
    const int lane = threadIdx.x;
    const int half = lane >> 4;
    const int lm   = lane & 15;

    if (lane < 8) sG[lane] = (lane < 7) ? g7[lane] : 0.f;

    const float* __restrict__ A = acc + (size_t)img * (HP * WP);
    for (int idx = lane; idx < 32 * 24; idx += 32) {
        const int r  = idx / 24, cc = idx % 24;
        const int gy = y0 - 3 + r, gx = x0 - 3 + cc;
        float v = 0.f;
        if (r < 22 && cc < 22 && (unsigned)gy < (unsigned)HP && (unsigned)gx < (unsigned)WP)
            v = A[gy * WP + gx];
        sIn[r][cc] = v;
    }
    __syncthreads();

    // ---- pass 1: horizontal band GEMM (two 16-row groups cover 22 rows) ----
    for (int grp = 0; grp < 2; ++grp) {
        const int rb = grp * 16;
        v8f d = {0.f, 0.f, 0.f, 0.f, 0.f, 0.f, 0.f, 0.f};
        #pragma unroll
        for (int s = 0; s < 6; ++s) {
            const int k0 = 4 * s + 2 * half;
            v2f a, bf;
            a.x = sIn[rb + lm][k0];
            a.y = sIn[rb + lm][k0 + 1];
            const int d0 = k0 - lm, d1 = d0 + 1;        // band: g[k-x], clamp -> sG[7]=0
            bf.x = sG[(unsigned)d0 < 7u ? d0 : 7];
            bf.y = sG[(unsigned)d1 < 7u ? d1 : 7];
            d = __builtin_amdgcn_wmma_f32_16x16x4_f32(false, a, false, bf,
                                                      (short)0, d, false, false);
        }
        #pragma unroll
        for (int r = 0; r < 8; ++r) {
            const int row = rb + r + 8 * half;
            if (row < 24) sT[row][lm] = d[r];
        }
    }
    __syncthreads();

    // ---- pass 2: vertical band GEMM ----
    v8f o = {0.f, 0.f, 0.f, 0.f, 0.f, 0.f, 0.f, 0.f};
    #pragma unroll
    for (int s = 0; s < 6; ++s) {
        const int k0 = 4 * s + 2 * half;
        v2f a, bf;
        const int d0 = k0 - lm, d1 = d0 + 1;            // band: g[k-y]
        a.x = sG[(unsigned)d0 < 7u ? d0 : 7];
        a.y = sG[(unsigned)d1 < 7u ? d1 : 7];
        bf.x = sT[k0][lm];
        bf.y = sT[k0 + 1][lm];
        o = __builtin_amdgcn_wmma_f32_16x16x4_f32(false, a, false, bf,
                                                  (short)0, o, false, false);
    }

    float* __restrict__ O = out + (size_t)img * (HP * WP);
    #pragma unroll
    for (int r = 0; r < 8; ++r) {
        const int row = y0 + r + 8 * half;
        O[row * WP + x0 + lm] = o[r];
    }
}

// ---------------------------------------------------------------------------
// Workspace layout (bytes):
//   [0,32)        g[8]
//   [32,2080)     fx[512]      [2080,4128)  fy[512]
//   [4128,6176)   nx[512]      [6176,8224)  ny[512]
//   [16384, ...)  acc: 64*192*192 floats = 9,437,184 B
// Total ~9.45 MB.
// ---------------------------------------------------------------------------
extern "C" void kernel_launch(void* const* d_in, const int* in_sizes, int n_in,
                              void* d_out, int out_size, void* d_ws, size_t ws_size,
                              hipStream_t stream) {
    const float* cube = (const float*)d_in[0];   // (16,128,128,128) f32
    const float* psf  = (const float*)d_in[1];   // (7,7) f32
    const float* dx   = (const float*)d_in[2];   // (4,128) f32
    const float* dy   = (const float*)d_in[3];   // (4,128) f32

    char* ws = (char*)d_ws;
    float* g   = (float*)(ws + 0);
    float* fx  = (float*)(ws + 32);
    float* fy  = (float*)(ws + 32 + 2048);
    int*   nx  = (int*)  (ws + 32 + 4096);
    int*   ny  = (int*)  (ws + 32 + 6144);
    float* acc = (float*)(ws + 16384);

    prep_kernel<<<1, 512, 0, stream>>>(psf, dx, dy, g, fx, fy, nx, ny);

    accum_kernel<<<dim3(NTILE * NTILE, NK, BATCH), 256, 0, stream>>>(
        cube, fx, fy, nx, ny, acc);

    conv_wmma_kernel<<<dim3(NTILE * NTILE, BATCH * NK), 32, 0, stream>>>(
        acc, g, (float*)d_out);
}